// Decoder_63402307223930
// MI455X (gfx1250) — compile-verified
//
#include <hip/hip_runtime.h>
#include <math.h>
#include <stdint.h>

// Problem constants (match reference)
static constexpr int VV = 32000;  // VOCAB
static constexpr int HH = 1280;   // HID
static constexpr int EE = 256;    // EMB
static constexpr int BB = 128;    // B
static constexpr int SS = 64;     // S
static constexpr int XW = HH + EE;   // 1536, width of x
static constexpr int G3 = 3 * HH;    // 3840

typedef __attribute__((ext_vector_type(2))) float v2f;
typedef __attribute__((ext_vector_type(8))) float v8f;

// ---------------------------------------------------------------------------
// K1: attention scores. scores[s*B+b] = dot(enc[s,b,:], w_e) + dot(h0[b,:], w_h) + attn_b
// ---------------------------------------------------------------------------
__global__ __launch_bounds__(256) void k_scores(const float* __restrict__ enc,
                                                const float* __restrict__ h0,
                                                const float* __restrict__ attn_w,
                                                const float* __restrict__ attn_b,
                                                float* __restrict__ scores) {
  const int lane = threadIdx.x & 31;
  const int wave = threadIdx.x >> 5;
  const int pair = blockIdx.x * 8 + wave;  // 0 .. S*B-1
  const int b = pair & (BB - 1);
  float acc = 0.0f;
  const float* erow = enc + (size_t)pair * HH;   // (s*B+b)*H
  const float* hrow = h0 + (size_t)b * HH;
  for (int i = lane; i < HH; i += 32) {
    acc += erow[i] * attn_w[HH + i];   // w_e
    acc += hrow[i] * attn_w[i];        // w_h
  }
#pragma unroll
  for (int off = 16; off > 0; off >>= 1) acc += __shfl_xor(acc, off, 32);
  if (lane == 0) scores[pair] = acc + attn_b[0];
}

// ---------------------------------------------------------------------------
// K2: softmax over the BATCH axis (axis=1 of (S,B)).  One block per s.
// ---------------------------------------------------------------------------
__global__ __launch_bounds__(128) void k_softmax_b(float* __restrict__ scores,
                                                   float* __restrict__ out_aw) {
  __shared__ float red[BB];
  const int s = blockIdx.x;
  const int b = threadIdx.x;
  float v = scores[s * BB + b];
  red[b] = v;
  __syncthreads();
#pragma unroll
  for (int off = 64; off > 0; off >>= 1) {
    if (b < off) red[b] = fmaxf(red[b], red[b + off]);
    __syncthreads();
  }
  const float m = red[0];
  __syncthreads();
  const float e = expf(v - m);
  red[b] = e;
  __syncthreads();
#pragma unroll
  for (int off = 64; off > 0; off >>= 1) {
    if (b < off) red[b] += red[b + off];
    __syncthreads();
  }
  const float r = e / red[0];
  scores[s * BB + b] = r;
  out_aw[s * BB + b] = r;
}

// ---------------------------------------------------------------------------
// K3: ctx (flat-reshape semantics!) + embedding gather + ReLU into x (B x 1536).
// ctx[b,h] = sum_s aw_flat[b*64+s] * enc_flat[(b*64+s)*H + h]
// ---------------------------------------------------------------------------
__global__ __launch_bounds__(256) void k_ctx_emb(const float* __restrict__ aw,
                                                 const float* __restrict__ enc,
                                                 const float* __restrict__ embedding,
                                                 const int* __restrict__ dec_in,
                                                 float* __restrict__ x) {
  __shared__ float w[SS];
  const int b = blockIdx.x;
  const int t = threadIdx.x;
  if (t < SS) w[t] = aw[b * SS + t];
  __syncthreads();
  float* xrow = x + (size_t)b * XW;
  const float* ebase = enc + (size_t)b * SS * HH;  // rows (b*64+s)
  for (int h = t; h < HH; h += 256) {
    float acc = 0.0f;
#pragma unroll 8
    for (int s = 0; s < SS; ++s) acc += w[s] * ebase[(size_t)s * HH + h];
    xrow[h] = fmaxf(acc, 0.0f);
  }
  const int row = dec_in[b];
  xrow[HH + t] = fmaxf(embedding[(size_t)row * EE + t], 0.0f);
}

// ---------------------------------------------------------------------------
// K4: WMMA f32 GEMM:  C[M=128, N] = A[128, K] * W[N, K]^T + bias[N]
//
// One wave owns a 16-column tile and ALL 8 row-tiles of M=128 (max W reuse:
// every W element is read from HBM exactly once).  The W stream is staged in
// 64-k chunks (16 rows x 64 floats = 4KB) into wave-private LDS using
// GLOBAL_LOAD_ASYNC_TO_LDS_B128 (ASYNCcnt), double-buffered: each wave keeps
// a full 4KB chunk in flight -> aggregate outstanding bytes match the
// HBM bandwidth-latency product, unlike per-fragment b64 loads.
// B fragments come from ds_load_b64 (pitch 68 floats => conflict-free banks);
// A fragments stay on global ping-pong loads (A panel is L2-resident).
//
// V_WMMA_F32_16X16X4_F32 operand mapping (ISA 7.12.2):
//   A (16x4):  lane m = lane&15, reg r -> k = kb + (lane>>4)*2 + r
//   B (4x16):  lane n = lane&15, reg r -> k = kb + (lane>>4)*2 + r
//   C (16x16): lane n = lane&15, reg r -> m = r + 8*(lane>>4)
// ---------------------------------------------------------------------------
static constexpr int KC    = 64;          // k-chunk size (16 wmma k-steps)
static constexpr int PITCH = 68;          // padded floats per LDS row
static constexpr int FBUF  = 16 * PITCH;  // floats per buffer (1088)

__global__ __launch_bounds__(256) void k_gemm_awt(const float* __restrict__ A, int lda,
                                                  const float* __restrict__ W,  // (N,K) rm
                                                  const float* __restrict__ bias,
                                                  float* __restrict__ C, int ldc,
                                                  int K) {
  __shared__ float ldsB[8 * 2 * FBUF];  // 69632 bytes
  const int lane = threadIdx.x & 31;
  const int wave = threadIdx.x >> 5;
  const int n0 = (blockIdx.x * 8 + wave) * 16;
  const int nl = lane & 15;
  const int half = lane >> 4;
  const int khalf = half * 2;

  v8f acc[8] = {};

#if defined(__gfx1250__)
  float* wbase0 = &ldsB[wave * 2 * FBUF];           // buffer 0 of this wave
  const int nch = K / KC;

  // ---- async stage of one 4KB chunk (8 x b128 per wave) --------------------
  // lane covers: row = 2*j + half (tile row 0..15), 4 floats at col (lane&15)*4
  auto stage = [&](int buf, int kc) {
    float* dstb = wbase0 + buf * FBUF;
#pragma unroll
    for (int j = 0; j < 8; ++j) {
      const int row = 2 * j + half;
      const float* g = W + (size_t)(n0 + row) * K + kc + nl * 4;
      const unsigned l = (unsigned)(uintptr_t)(dstb + row * PITCH + nl * 4);
      asm volatile("global_load_async_to_lds_b128 %0, %1, off"
                   :: "v"(l), "v"(g) : "memory");
    }
  };

  stage(0, 0);
  stage(1, KC);

  // A fragment prologue (k = 0)
  const float* Ap = A + (size_t)nl * lda + khalf;
  v2f af_c[8];
#pragma unroll
  for (int i = 0; i < 8; ++i) af_c[i] = *(const v2f*)(Ap + (size_t)i * 16 * lda);

  for (int c = 0; c < nch; ++c) {
    // wait for chunk c: allow the 8 async ops of chunk c+1 to stay in flight
    if (c + 2 <= nch) {
      asm volatile("s_wait_asynccnt 8" ::: "memory");
    } else {
      asm volatile("s_wait_asynccnt 0" ::: "memory");
    }
    const float* bufp = wbase0 + (c & 1) * FBUF + nl * PITCH + khalf;

#pragma unroll
    for (int step = 0; step < 16; ++step) {
      const v2f bf = *(const v2f*)(bufp + step * 4);   // ds_load_b64
      // prefetch next k-step's A fragments (clamped at the end; redundant ok)
      int kn = c * KC + step * 4 + 4;
      if (kn > K - 4) kn = K - 4;
      v2f af_n[8];
#pragma unroll
      for (int i = 0; i < 8; ++i)
        af_n[i] = *(const v2f*)(Ap + (size_t)i * 16 * lda + kn);
#pragma unroll
      for (int i = 0; i < 8; ++i)
        acc[i] = __builtin_amdgcn_wmma_f32_16x16x4_f32(false, af_c[i], false, bf,
                                                       (short)0, acc[i], false, false);
#pragma unroll
      for (int i = 0; i < 8; ++i) af_c[i] = af_n[i];
    }

    // restage the buffer we just drained with chunk c+2
    if (c + 2 < nch) stage(c & 1, (c + 2) * KC);
  }
#else
  (void)lda; (void)K; (void)khalf; (void)half;  // host pass: never executed
#endif

  const float bv = bias ? bias[n0 + nl] : 0.0f;
  const int mbase = half * 8;
#pragma unroll
  for (int mt = 0; mt < 8; ++mt) {
#pragma unroll
    for (int r = 0; r < 8; ++r) {
      const int m = mt * 16 + mbase + r;
      C[(size_t)m * ldc + (n0 + nl)] = acc[mt][r] + bv;
    }
  }
}

// ---------------------------------------------------------------------------
// K5: GRU gates -> h_new (written straight into d_out h slot)
// ---------------------------------------------------------------------------
__global__ __launch_bounds__(256) void k_gru(const float* __restrict__ gi,
                                             const float* __restrict__ gh,
                                             const float* __restrict__ h0,
                                             float* __restrict__ h_out) {
  const int i = blockIdx.x * 256 + threadIdx.x;  // 0 .. B*H-1
  const int b = i / HH;
  const int h = i - b * HH;
  const float* gib = gi + (size_t)b * G3;
  const float* ghb = gh + (size_t)b * G3;
  const float r = 1.0f / (1.0f + expf(-(gib[h] + ghb[h])));
  const float z = 1.0f / (1.0f + expf(-(gib[HH + h] + ghb[HH + h])));
  const float n = tanhf(gib[2 * HH + h] + r * ghb[2 * HH + h]);
  h_out[i] = (1.0f - z) * n + z * h0[i];
}

// ---------------------------------------------------------------------------
// K6: row-wise log_softmax over VOCAB.  One block per batch row.
// ---------------------------------------------------------------------------
__global__ __launch_bounds__(256) void k_logsoftmax(const float* __restrict__ logits,
                                                    float* __restrict__ outp) {
  __shared__ float red[256];
  const int b = blockIdx.x;
  const int t = threadIdx.x;
  const float* row = logits + (size_t)b * VV;
  float m = -INFINITY;
  for (int v = t; v < VV; v += 256) m = fmaxf(m, row[v]);
  red[t] = m;
  __syncthreads();
#pragma unroll
  for (int off = 128; off > 0; off >>= 1) {
    if (t < off) red[t] = fmaxf(red[t], red[t + off]);
    __syncthreads();
  }
  m = red[0];
  __syncthreads();
  float s = 0.0f;
  for (int v = t; v < VV; v += 256) s += expf(row[v] - m);
  red[t] = s;
  __syncthreads();
#pragma unroll
  for (int off = 128; off > 0; off >>= 1) {
    if (t < off) red[t] += red[t + off];
    __syncthreads();
  }
  const float lse = m + logf(red[0]);
  float* orow = outp + (size_t)b * VV;
  for (int v = t; v < VV; v += 256) orow[v] = row[v] - lse;
}

// ---------------------------------------------------------------------------
extern "C" void kernel_launch(void* const* d_in, const int* in_sizes, int n_in,
                              void* d_out, int out_size, void* d_ws, size_t ws_size,
                              hipStream_t stream) {
  (void)in_sizes; (void)n_in; (void)out_size; (void)ws_size;
  const int*   dec_in    = (const int*)d_in[0];
  const float* h0        = (const float*)d_in[1];   // (1,B,H) -> (B,H)
  const float* enc       = (const float*)d_in[2];   // (S,B,H)
  const float* embedding = (const float*)d_in[3];   // (V,E)
  const float* attn_w    = (const float*)d_in[4];   // (1,2H)
  const float* attn_b    = (const float*)d_in[5];   // (1,)
  const float* w_ih      = (const float*)d_in[6];   // (3H, H+E)
  const float* w_hh      = (const float*)d_in[7];   // (3H, H)
  const float* b_ih      = (const float*)d_in[8];
  const float* b_hh      = (const float*)d_in[9];
  const float* dense_w   = (const float*)d_in[10];  // (V, H)
  const float* dense_b   = (const float*)d_in[11];

  float* out      = (float*)d_out;
  float* out_logp = out;                               // B*V
  float* out_h    = out + (size_t)BB * VV;             // B*H
  float* out_aw   = out_h + (size_t)BB * HH;           // S*B

  float* ws      = (float*)d_ws;
  float* scores  = ws;                                 // S*B          (becomes aw)
  float* x       = scores + SS * BB;                   // B*1536
  float* gi      = x + (size_t)BB * XW;                // B*3840
  float* gh      = gi + (size_t)BB * G3;               // B*3840
  float* logits  = gh + (size_t)BB * G3;               // B*V

  // 1) attention scores: S*B/8 = 1024 blocks, 8 waves each
  k_scores<<<(SS * BB) / 8, 256, 0, stream>>>(enc, h0, attn_w, attn_b, scores);
  // 2) softmax over batch axis
  k_softmax_b<<<SS, BB, 0, stream>>>(scores, out_aw);
  // 3) ctx + embedding + relu -> x
  k_ctx_emb<<<BB, 256, 0, stream>>>(scores, enc, embedding, dec_in, x);
  // 4) gate GEMMs (WMMA): gi = x @ w_ih^T + b_ih ; gh = h0 @ w_hh^T + b_hh
  k_gemm_awt<<<G3 / 128, 256, 0, stream>>>(x, XW, w_ih, b_ih, gi, G3, XW);
  k_gemm_awt<<<G3 / 128, 256, 0, stream>>>(h0, HH, w_hh, b_hh, gh, G3, HH);
  // 5) GRU elementwise -> h_new (directly into d_out)
  k_gru<<<(BB * HH) / 256, 256, 0, stream>>>(gi, gh, h0, out_h);
  // 6) logits = h_new @ dense_w^T + dense_b  (WMMA, 250 blocks x 8 waves)
  k_gemm_awt<<<VV / 128, 256, 0, stream>>>(out_h, HH, dense_w, dense_b, logits, VV, HH);
  // 7) log_softmax per row
  k_logsoftmax<<<BB, 256, 0, stream>>>(logits, out_logp);
}